// GNN_360777253507
// MI455X (gfx1250) — compile-verified
//
#include <hip/hip_runtime.h>

// MI455X / gfx1250, wave32. Matrix math via V_WMMA_F32_16X16X4_F32 (exact f32).
typedef float v2f __attribute__((ext_vector_type(2)));
typedef float v8f __attribute__((ext_vector_type(8)));

#define NCLS 10

// ---------------------------------------------------------------------------
// Scatter-add of edge messages: agg[dst] += feat[src] * ew   (D = 32 or 64)
// One thread per (edge, 4-float quad). Atomics land in L2 (agg fits in 192MB L2).
// ---------------------------------------------------------------------------
__global__ void scatter_edges(const float* __restrict__ feat,      // [N, D]
                              const long long* __restrict__ ei,    // [2, E]
                              const float* __restrict__ ew,        // [E]
                              float* __restrict__ agg,             // [N, D]
                              int E, int D, int qshift)            // qshift = log2(D/4)
{
    long long tid = (long long)blockIdx.x * blockDim.x + threadIdx.x;
    if (tid >= ((long long)E << qshift)) return;
    int e = (int)(tid >> qshift);
    int q = (int)(tid & ((1 << qshift) - 1));
    int s = (int)ei[e];
    int d = (int)ei[(size_t)E + e];
    float w = ew[e];
    const float4 v = *(const float4*)(feat + (size_t)s * D + q * 4);
    float* ap = agg + (size_t)d * D + q * 4;
    __hip_atomic_fetch_add(ap + 0, v.x * w, __ATOMIC_RELAXED, __HIP_MEMORY_SCOPE_AGENT);
    __hip_atomic_fetch_add(ap + 1, v.y * w, __ATOMIC_RELAXED, __HIP_MEMORY_SCOPE_AGENT);
    __hip_atomic_fetch_add(ap + 2, v.z * w, __ATOMIC_RELAXED, __HIP_MEMORY_SCOPE_AGENT);
    __hip_atomic_fetch_add(ap + 3, v.w * w, __ATOMIC_RELAXED, __HIP_MEMORY_SCOPE_AGENT);
}

// ---------------------------------------------------------------------------
// Layer 1 node update: h1 = relu( [agg1 | x] @ [W1_rel ; W1_root]^T + b1 )
// One wave per 16-node tile; 4 output tiles of 16; K = 64 in steps of 4.
// A layout (16x4 f32): lane m holds row m; half 0 -> K pair {0,1}, half 1 -> {2,3}.
// B layout (4x16):     vgpr i, half h -> K = h*2+i, N = lane%16.
// ---------------------------------------------------------------------------
__global__ void __launch_bounds__(256)
node_update1(const float* __restrict__ agg1,  // [N,32]
             const float* __restrict__ x,     // [N,32]
             const float* __restrict__ Wrel,  // [64,32]
             const float* __restrict__ Wroot, // [64,32]
             const float* __restrict__ bias,  // [64]
             float* __restrict__ h1,          // [N,64]
             int N, int numTiles)
{
    int lane = threadIdx.x & 31;
    int wid  = threadIdx.x >> 5;
    int tile = blockIdx.x * 8 + wid;
    if (tile >= numTiles) return;
    int m    = lane & 15;
    int half = lane >> 4;

    int rowA = tile * 16 + m;
    if (rowA >= N) rowA = N - 1;               // tail-safe (stores are guarded)
    const float* arow = agg1 + (size_t)rowA * 32;
    const float* xrow = x    + (size_t)rowA * 32;

    v8f acc[4] = {};
#pragma unroll
    for (int s = 0; s < 16; ++s) {             // K = 64 total, 4 per wmma
        int kk = s * 4 + half * 2;
        v2f a = (s < 8) ? *(const v2f*)(arow + kk)
                        : *(const v2f*)(xrow + (kk - 32));
#pragma unroll
        for (int t = 0; t < 4; ++t) {
            const float* wr = (s < 8) ? (Wrel  + (size_t)(t * 16 + m) * 32 + kk)
                                      : (Wroot + (size_t)(t * 16 + m) * 32 + (kk - 32));
            v2f b = *(const v2f*)wr;
            acc[t] = __builtin_amdgcn_wmma_f32_16x16x4_f32(
                false, a, false, b, (short)0, acc[t], false, false);
        }
    }

    // D layout: vgpr j, half h -> node row = tile*16 + h*8 + j, col = outBase + m
#pragma unroll
    for (int t = 0; t < 4; ++t) {
        float bv = bias[t * 16 + m];
#pragma unroll
        for (int j = 0; j < 8; ++j) {
            int node = tile * 16 + half * 8 + j;
            if (node < N) {
                float v = acc[t][j] + bv;
                h1[(size_t)node * 64 + t * 16 + m] = v > 0.f ? v : 0.f;
            }
        }
    }
}

// ---------------------------------------------------------------------------
// Layer 2 node update fused with final FC:
//   h2  = relu( [agg2 | h1] @ [W2_rel ; W2_root]^T + b2 )   (WMMA, K=128)
//   out = h2 @ fc_w^T + fc_b                                 (via padded LDS)
// Wave-local LDS tile + s_wait_dscnt: no workgroup barrier -> tail waves safe.
// ---------------------------------------------------------------------------
__global__ void __launch_bounds__(256)
node_update2_fc(const float* __restrict__ agg2,  // [N,64]
                const float* __restrict__ h1,    // [N,64]
                const float* __restrict__ Wrel,  // [64,64]
                const float* __restrict__ Wroot, // [64,64]
                const float* __restrict__ bias,  // [64]
                const float* __restrict__ fcw,   // [10,64]
                const float* __restrict__ fcb,   // [10]
                float* __restrict__ out,         // [N,10]
                int N, int numTiles)
{
    __shared__ float lds[8 * 16 * 65];           // 8 waves x (16 nodes x 65-padded)
    int lane = threadIdx.x & 31;
    int wid  = threadIdx.x >> 5;
    float* myl = lds + wid * 16 * 65;
    int tile = blockIdx.x * 8 + wid;
    if (tile >= numTiles) return;
    int m    = lane & 15;
    int half = lane >> 4;

    int rowA = tile * 16 + m;
    if (rowA >= N) rowA = N - 1;
    const float* arow = agg2 + (size_t)rowA * 64;
    const float* hrow = h1   + (size_t)rowA * 64;

    v8f acc[4] = {};
#pragma unroll
    for (int s = 0; s < 32; ++s) {               // K = 128 total
        int kk = s * 4 + half * 2;
        v2f a = (s < 16) ? *(const v2f*)(arow + kk)
                         : *(const v2f*)(hrow + (kk - 64));
#pragma unroll
        for (int t = 0; t < 4; ++t) {
            const float* wr = (s < 16) ? (Wrel  + (size_t)(t * 16 + m) * 64 + kk)
                                       : (Wroot + (size_t)(t * 16 + m) * 64 + (kk - 64));
            v2f b = *(const v2f*)wr;
            acc[t] = __builtin_amdgcn_wmma_f32_16x16x4_f32(
                false, a, false, b, (short)0, acc[t], false, false);
        }
    }

    // ReLU + stash h2 tile into LDS (row stride 65 -> conflict-free column reads)
#pragma unroll
    for (int t = 0; t < 4; ++t) {
        float bv = bias[t * 16 + m];
#pragma unroll
        for (int j = 0; j < 8; ++j) {
            float v = acc[t][j] + bv;
            myl[(half * 8 + j) * 65 + t * 16 + m] = v > 0.f ? v : 0.f;
        }
    }
    asm volatile("s_wait_dscnt 0" ::: "memory"); // wave-local LDS WAR/RAW fence

    // FC 64 -> 10: lane handles node (lane&15), classes [half*5, half*5+5)
    int nodeL = lane & 15;
    int node  = tile * 16 + nodeL;
    const float* hv = myl + nodeL * 65;
#pragma unroll
    for (int c5 = 0; c5 < 5; ++c5) {
        int cc = half * 5 + c5;
        float a0 = fcb[cc];
#pragma unroll
        for (int h = 0; h < 64; ++h)
            a0 = fmaf(hv[h], fcw[cc * 64 + h], a0);
        if (node < N) out[(size_t)node * NCLS + cc] = a0;
    }
}

// ---------------------------------------------------------------------------
extern "C" void kernel_launch(void* const* d_in, const int* in_sizes, int n_in,
                              void* d_out, int out_size, void* d_ws, size_t ws_size,
                              hipStream_t stream)
{
    const float*     x       = (const float*)d_in[0];      // [N,32]
    const long long* ei      = (const long long*)d_in[1];  // [2,E] int64
    const float*     ea      = (const float*)d_in[2];      // [E]
    const float*     W1_rel  = (const float*)d_in[3];      // [64,32]
    const float*     b1      = (const float*)d_in[4];      // [64]
    const float*     W1_root = (const float*)d_in[5];      // [64,32]
    const float*     W2_rel  = (const float*)d_in[6];      // [64,64]
    const float*     b2      = (const float*)d_in[7];      // [64]
    const float*     W2_root = (const float*)d_in[8];      // [64,64]
    const float*     fcw     = (const float*)d_in[9];      // [10,64]
    const float*     fcb     = (const float*)d_in[10];     // [10]
    float*           out     = (float*)d_out;

    const int N = in_sizes[0] / 32;
    const int E = in_sizes[2];

    float* agg1 = (float*)d_ws;                 // [N,32]
    float* h1   = agg1 + (size_t)N * 32;        // [N,64]
    float* agg2 = h1   + (size_t)N * 64;        // [N,64]

    // Zero accumulators (graph-capture-safe async memset).
    hipMemsetAsync(d_ws, 0, (size_t)N * (32 + 64 + 64) * sizeof(float), stream);

    // Layer 1 scatter: E * 8 quads
    {
        long long total = (long long)E << 3;
        int blocks = (int)((total + 255) / 256);
        scatter_edges<<<blocks, 256, 0, stream>>>(x, ei, ea, agg1, E, 32, 3);
    }

    int tiles   = (N + 15) / 16;
    int nblocks = (tiles + 7) / 8;
    node_update1<<<nblocks, 256, 0, stream>>>(agg1, x, W1_rel, W1_root, b1, h1, N, tiles);

    // Layer 2 scatter: E * 16 quads
    {
        long long total = (long long)E << 4;
        int blocks = (int)((total + 255) / 256);
        scatter_edges<<<blocks, 256, 0, stream>>>(h1, ei, ea, agg2, E, 64, 4);
    }

    node_update2_fc<<<nblocks, 256, 0, stream>>>(agg2, h1, W2_rel, W2_root, b2,
                                                 fcw, fcb, out, N, tiles);
}